// piSGC_39135742001431
// MI455X (gfx1250) — compile-verified
//
#include <hip/hip_runtime.h>
#include <math.h>

#define ALPHA 0.05f
#define FDIM 64
#define CPAD 48   // C=40 padded to 3 WMMA N-tiles

typedef float v8f __attribute__((ext_vector_type(8)));
typedef float v2f __attribute__((ext_vector_type(2)));

// ---------------- zero init (float4) ----------------
__global__ void k_zero4(float4* __restrict__ p, long long n4) {
    long long i = (long long)blockIdx.x * blockDim.x + threadIdx.x;
    if (i < n4) p[i] = make_float4(0.f, 0.f, 0.f, 0.f);
}

// ---------------- degree scatter ----------------
__global__ void k_deg(const float* __restrict__ ea, const long long* __restrict__ col,
                      float* __restrict__ deg, long long E) {
    long long e = (long long)blockIdx.x * blockDim.x + threadIdx.x;
    if (e >= E) return;
    __hip_atomic_fetch_add(&deg[col[e]], ea[e], __ATOMIC_RELAXED, __HIP_MEMORY_SCOPE_AGENT);
}

// ---------------- deg -> dis (in place) ----------------
__global__ void k_dis(float* __restrict__ deg, long long N) {
    long long i = (long long)blockIdx.x * blockDim.x + threadIdx.x;
    if (i >= N) return;
    float d = deg[i];
    deg[i] = (d > 0.f) ? rsqrtf(fmaxf(d, 1e-12f)) : 0.f;
}

// ---------------- per-edge norm ----------------
__global__ void k_norm(const float* __restrict__ dis, const float* __restrict__ ea,
                       const long long* __restrict__ row, const long long* __restrict__ col,
                       float* __restrict__ norm, long long E) {
    long long e = (long long)blockIdx.x * blockDim.x + threadIdx.x;
    if (e >= E) return;
    norm[e] = dis[row[e]] * ea[e] * dis[col[e]];
}

// ---------------- sparse conv: hout[col] += norm * hin[row] ----------------
// 16 threads per edge, float4 per thread (64 floats / edge)
__global__ void k_conv(const float* __restrict__ hin, const long long* __restrict__ row,
                       const long long* __restrict__ col, const float* __restrict__ norm,
                       float* __restrict__ hout, long long E) {
    int t = threadIdx.x & 15;
    long long e = ((long long)blockIdx.x * blockDim.x + threadIdx.x) >> 4;
    if (e >= E) return;
    long long r = row[e], c = col[e];
    float w = norm[e];
    const float4* src = (const float4*)(hin + r * FDIM);
    float4 v = src[t];
    float* dst = hout + c * FDIM + t * 4;
    __hip_atomic_fetch_add(dst + 0, w * v.x, __ATOMIC_RELAXED, __HIP_MEMORY_SCOPE_AGENT);
    __hip_atomic_fetch_add(dst + 1, w * v.y, __ATOMIC_RELAXED, __HIP_MEMORY_SCOPE_AGENT);
    __hip_atomic_fetch_add(dst + 2, w * v.z, __ATOMIC_RELAXED, __HIP_MEMORY_SCOPE_AGENT);
    __hip_atomic_fetch_add(dst + 3, w * v.w, __ATOMIC_RELAXED, __HIP_MEMORY_SCOPE_AGENT);
}

// ---------------- feat = alpha*x + scale*(h1+h2), written into h1 ----------------
__global__ void k_combine(const float4* __restrict__ x, float4* __restrict__ h1,
                          const float4* __restrict__ h2, float scale, long long n4) {
    long long i = (long long)blockIdx.x * blockDim.x + threadIdx.x;
    if (i >= n4) return;
    float4 a = x[i], b = h1[i], c = h2[i];
    float4 o;
    o.x = ALPHA * a.x + scale * (b.x + c.x);
    o.y = ALPHA * a.y + scale * (b.y + c.y);
    o.z = ALPHA * a.z + scale * (b.z + c.z);
    o.w = ALPHA * a.w + scale * (b.w + c.w);
    h1[i] = o;
}

// ---------------- pad W/bias into guard-free WMMA-friendly layout ----------------
// Wp[pk*CPAD + c] = ( W[2pk][c], W[2pk+1][c] ), pk = 0..31, zero-padded for c >= C.
// bp[c] = bias[c] (zero-padded).
__global__ void k_padw(const float* __restrict__ W, const float* __restrict__ b,
                       float2* __restrict__ Wp, float* __restrict__ bp, int C) {
    int i = blockIdx.x * blockDim.x + threadIdx.x;
    if (i < 32 * CPAD) {
        int pk = i / CPAD, c = i % CPAD;
        float2 v = make_float2(0.f, 0.f);
        if (c < C) { v.x = W[(2 * pk) * C + c]; v.y = W[(2 * pk + 1) * C + c]; }
        Wp[i] = v;
    } else if (i < 32 * CPAD + CPAD) {
        int c = i - 32 * CPAD;
        bp[c] = (c < C) ? b[c] : 0.f;
    }
}

// ---------------- GEMM via V_WMMA_F32_16X16X4_F32 (guard-free inner loop) ----------------
// A = feat [M x 64], Wp = packed padded weights, bp = padded bias.
// block = 96 threads (3 waves); wave w handles N-tile w of M-tile blockIdx.x.
// K = 64 -> 16 WMMA steps of K=4. A frag (32b A 16x4): lanes 0-15 v0=K0,v1=K1;
// lanes 16-31 v0=K2,v1=K3 -> per-step contiguous pair at float2 index pk = 2k0+half.
// C/D: vgpr r, lanes 0-15 -> (M=r, N=lane); lanes 16-31 -> (M=8+r, N=lane-16).
__global__ void k_gemm_wmma(const float* __restrict__ A, const float2* __restrict__ Wp,
                            const float* __restrict__ bp, float* __restrict__ out,
                            int M, int C) {
    int wave = threadIdx.x >> 5;   // n-tile 0..2
    int lane = threadIdx.x & 31;
    int half = lane >> 4;
    int l = lane & 15;
    int m0 = blockIdx.x * 16;
    int n0 = wave * 16;
    int cidx = n0 + l;

    float bval = bp[cidx];
    v8f acc;
#pragma unroll
    for (int i = 0; i < 8; ++i) acc[i] = bval;

    int mrow = min(m0 + l, M - 1);                       // clamp, no branch
    const v2f* arow = (const v2f*)(A + (long long)mrow * FDIM);
    const v2f* wcol = (const v2f*)Wp;

#pragma unroll
    for (int k0 = 0; k0 < 16; ++k0) {
        int pk = k0 * 2 + half;
        v2f a = arow[pk];                                // global_load_b64
        v2f b = wcol[pk * CPAD + cidx];                  // global_load_b64
        acc = __builtin_amdgcn_wmma_f32_16x16x4_f32(false, a, false, b,
                                                    (short)0, acc, false, false);
    }

    if (cidx < C) {
#pragma unroll
        for (int r = 0; r < 8; ++r) {
            int orow = m0 + half * 8 + r;
            if (orow < M) out[(long long)orow * C + cidx] = acc[r];
        }
    }
}

// ---------------- in-place row log_softmax, one wave per row ----------------
__global__ void k_logsoftmax(float* __restrict__ out, int N, int C) {
    int wavesPerBlock = blockDim.x >> 5;
    int waveId = threadIdx.x >> 5;
    int lane = threadIdx.x & 31;
    long long rowId = (long long)blockIdx.x * wavesPerBlock + waveId;
    if (rowId >= N) return;
    float* rowp = out + rowId * C;
    bool v0ok = (lane < C);
    bool v1ok = (lane + 32 < C);
    float v0 = v0ok ? rowp[lane] : -__builtin_inff();
    float v1 = v1ok ? rowp[lane + 32] : -__builtin_inff();
    float m = fmaxf(v0, v1);
#pragma unroll
    for (int off = 16; off > 0; off >>= 1) m = fmaxf(m, __shfl_xor(m, off, 32));
    float s = (v0ok ? __expf(v0 - m) : 0.f) + (v1ok ? __expf(v1 - m) : 0.f);
#pragma unroll
    for (int off = 16; off > 0; off >>= 1) s += __shfl_xor(s, off, 32);
    float lse = m + __logf(s);
    if (v0ok) rowp[lane] = v0 - lse;
    if (v1ok) rowp[lane + 32] = v1 - lse;
}

extern "C" void kernel_launch(void* const* d_in, const int* in_sizes, int n_in,
                              void* d_out, int out_size, void* d_ws, size_t ws_size,
                              hipStream_t stream) {
    const float* x      = (const float*)d_in[0];       // [N,64]
    const float* ea     = (const float*)d_in[1];       // [E]
    const float* lin_w  = (const float*)d_in[2];       // [64,C]
    const float* lin_b  = (const float*)d_in[3];       // [C]
    const long long* ei = (const long long*)d_in[4];   // [2,E] int64

    long long N = in_sizes[0] / FDIM;
    long long E = in_sizes[1];
    int C = in_sizes[3];

    const long long* row = ei;
    const long long* col = ei + E;

    // workspace layout (floats): deg[N] | h1[N*64] | h2[N*64] | norm[E] | Wp[32*CPAD*2] | bp[CPAD]
    float* deg  = (float*)d_ws;
    float* h1   = deg + N;
    float* h2   = h1 + N * FDIM;
    float* norm = h2 + N * FDIM;
    float* wp   = norm + E;                 // 8-byte aligned (even float offset)
    float* bp   = wp + 32 * CPAD * 2;
    float* logits = (float*)d_out;

    const int B = 256;

    // 1. zero deg + h1 + h2 (contiguous N*(1+128) floats; divisible by 4)
    long long zc4 = (N * (1 + 2 * FDIM)) / 4;
    k_zero4<<<(int)((zc4 + B - 1) / B), B, 0, stream>>>((float4*)deg, zc4);

    // 2. degree scatter
    k_deg<<<(int)((E + B - 1) / B), B, 0, stream>>>(ea, col, deg, E);

    // 3. dis
    k_dis<<<(int)((N + B - 1) / B), B, 0, stream>>>(deg, N);

    // 4. per-edge norm
    k_norm<<<(int)((E + B - 1) / B), B, 0, stream>>>(deg, ea, row, col, norm, E);

    // 5. conv layer 1: h1 = conv(x); conv layer 2: h2 = conv(h1)
    long long convThreads = E * 16;
    int convBlocks = (int)((convThreads + B - 1) / B);
    k_conv<<<convBlocks, B, 0, stream>>>(x, row, col, norm, h1, E);
    k_conv<<<convBlocks, B, 0, stream>>>(h1, row, col, norm, h2, E);

    // 6. feat = ALPHA*x + scale*(h1+h2)  -> h1
    float scale = (1.f - ALPHA) / 2.f;
    long long n4 = (N * FDIM) / 4;
    k_combine<<<(int)((n4 + B - 1) / B), B, 0, stream>>>((const float4*)x, (float4*)h1,
                                                         (const float4*)h2, scale, n4);

    // 7. pad/pack weights + bias (tiny)
    int padThreads = 32 * CPAD + CPAD;
    k_padw<<<(padThreads + B - 1) / B, B, 0, stream>>>(lin_w, lin_b, (float2*)wp, bp, C);

    // 8. logits = feat @ lin_w + lin_b via fp32 WMMA (3 N-tiles cover C<=48)
    int mTiles = (int)((N + 15) / 16);
    k_gemm_wmma<<<mTiles, 96, 0, stream>>>(h1, (const float2*)wp, bp, logits, (int)N, C);

    // 9. in-place log_softmax, 8 rows (waves) per block
    int rowsPerBlock = B / 32;
    k_logsoftmax<<<(int)((N + rowsPerBlock - 1) / rowsPerBlock), B, 0, stream>>>(logits, (int)N, C);
}